// GCN_26792005993066
// MI455X (gfx1250) — compile-verified
//
#include <hip/hip_runtime.h>
#include <hip/hip_bf16.h>
#include <math.h>

#define N_NODES   100000   // multiple of 16 -> all M-tiles full
#define IN_FEATS  64
#define N_HIDDEN  64
#define N_CLASSES 40

typedef __attribute__((ext_vector_type(2))) float v2f;
typedef __attribute__((ext_vector_type(8))) float v8f;

// ---------------------------------------------------------------------------
// Zero-fill a float buffer (grid-stride).
// ---------------------------------------------------------------------------
__global__ void gcn_zero_kernel(float* __restrict__ p, long long n) {
    long long i = (long long)blockIdx.x * blockDim.x + threadIdx.x;
    long long stride = (long long)gridDim.x * blockDim.x;
    for (; i < n; i += stride) p[i] = 0.0f;
}

// ---------------------------------------------------------------------------
// Edge-parallel scatter-add aggregation: agg[dst] += h[src], feature dim = 64.
// Each thread moves one float4 (4 feats) of one edge: nEdges*16 threads.
// Atomic adds resolve in L2 (both buffers fit in the 192MB L2).
// ---------------------------------------------------------------------------
__global__ void gcn_scatter_add_kernel(const float* __restrict__ h,
                                       const int* __restrict__ src,
                                       const int* __restrict__ dst,
                                       float* __restrict__ agg,
                                       int nEdges) {
    long long idx = (long long)blockIdx.x * blockDim.x + threadIdx.x;
    long long total = (long long)nEdges * 16;
    if (idx >= total) return;
    int e  = (int)(idx >> 4);
    int f4 = ((int)idx & 15) * 4;
    int s = src[e];
    int d = dst[e];
    const float4 v = *(const float4*)(h + (size_t)s * 64 + f4);
    float* out = agg + (size_t)d * 64 + f4;
    atomicAdd(out + 0, v.x);
    atomicAdd(out + 1, v.y);
    atomicAdd(out + 2, v.z);
    atomicAdd(out + 3, v.w);
}

// ---------------------------------------------------------------------------
// Y[nrows x NCOLS] = X[nrows x 64] @ W[64 x NCOLS] + bias, optional ReLU.
// fp32 WMMA: V_WMMA_F32_16X16X4_F32. One wave per 16-row M-tile.
// nrows must be a multiple of 16 (N_NODES is), so no row guards -> stores
// are straight-line (no exec-mask branches). Column guard folds away for
// full 16-wide N tiles.
//
// A-frag (16x4 f32): lane 0-15 -> row M=lane, {v0,v1}={K0,K1};
//                    lane 16-31 -> row M=lane-16, {v0,v1}={K2,K3}.
// B-frag (4x16 f32): lane 0-15 -> col N=lane, {v0,v1}=rows{K0,K1};
//                    lane 16-31 -> col N=lane-16, {v0,v1}=rows{K2,K3}.
// C/D (16x16 f32):   lane 0-15 -> N=lane, VGPR i -> M=i;
//                    lane 16-31 -> N=lane-16, VGPR i -> M=i+8.
// ---------------------------------------------------------------------------
template <int NCOLS, bool RELU>
__global__ void gcn_gemm_wmma_kernel(const float* __restrict__ X,
                                     const float* __restrict__ W,
                                     const float* __restrict__ bias,
                                     float* __restrict__ Y,
                                     int nrows) {
    constexpr int NT = (NCOLS + 15) / 16;  // N-tiles of 16
    const int lane  = threadIdx.x & 31;
    const int wave  = threadIdx.x >> 5;
    const int mtile = blockIdx.x * (blockDim.x >> 5) + wave;
    const int m0    = mtile * 16;
    if (m0 >= nrows) return;  // wave-uniform; EXEC stays full for WMMA

    const int ml  = lane & 15;        // row-in-tile (A) / col-in-tile (B,D)
    const int kb  = (lane >> 4) * 2;  // K sub-base from lane half
    const int row = m0 + ml;

    // Preload all 16 K-step A fragments as 8-byte vector loads.
    const float* __restrict__ xrow = X + (size_t)row * 64 + kb;
    v2f afrag[16];
#pragma unroll
    for (int kk = 0; kk < 16; ++kk)
        afrag[kk] = *(const v2f*)(xrow + kk * 4);

#pragma unroll
    for (int nt = 0; nt < NT; ++nt) {
        const int  n     = nt * 16 + ml;
        // Folds to compile-time true for full tiles (all tiles when NCOLS%16==0).
        const bool colOK = ((nt + 1) * 16 <= NCOLS) || (n < NCOLS);
        v8f c = {};
#pragma unroll
        for (int kk = 0; kk < 16; ++kk) {
            const int k0 = kk * 4 + kb;
            v2f bfrag;
            bfrag[0] = colOK ? W[(size_t)(k0 + 0) * NCOLS + n] : 0.0f;
            bfrag[1] = colOK ? W[(size_t)(k0 + 1) * NCOLS + n] : 0.0f;
            c = __builtin_amdgcn_wmma_f32_16x16x4_f32(
                    /*neg_a=*/false, afrag[kk],
                    /*neg_b=*/false, bfrag,
                    /*c_mod=*/(short)0, c,
                    /*reuse_a=*/false, /*reuse_b=*/false);
        }
        const float bv = colOK ? bias[n] : 0.0f;
        const int   mh = (lane >> 4) * 8;
        float* __restrict__ yp = Y + (size_t)(m0 + mh) * NCOLS + n;
#pragma unroll
        for (int i = 0; i < 8; ++i) {
            float v = c[i] + bv;
            if (RELU) v = v > 0.0f ? v : 0.0f;
            if (colOK) yp[(size_t)i * NCOLS] = v;
        }
    }
}

// ---------------------------------------------------------------------------
// Per-class (column) max and log-sum-exp over all nodes (axis=0 softmax).
// One block per class, strided over rows.
// ---------------------------------------------------------------------------
__global__ void gcn_col_logsumexp_kernel(const float* __restrict__ rep,
                                         int nrows,
                                         float* __restrict__ colmax,
                                         float* __restrict__ colse) {
    const int c = blockIdx.x;
    __shared__ float sdata[256];

    float m = -INFINITY;
    for (int i = threadIdx.x; i < nrows; i += blockDim.x)
        m = fmaxf(m, rep[(size_t)i * N_CLASSES + c]);
    sdata[threadIdx.x] = m;
    __syncthreads();
    for (int s = 128; s > 0; s >>= 1) {
        if (threadIdx.x < s)
            sdata[threadIdx.x] = fmaxf(sdata[threadIdx.x], sdata[threadIdx.x + s]);
        __syncthreads();
    }
    const float mx = sdata[0];
    __syncthreads();

    float sum = 0.0f;
    for (int i = threadIdx.x; i < nrows; i += blockDim.x)
        sum += expf(rep[(size_t)i * N_CLASSES + c] - mx);
    sdata[threadIdx.x] = sum;
    __syncthreads();
    for (int s = 128; s > 0; s >>= 1) {
        if (threadIdx.x < s)
            sdata[threadIdx.x] += sdata[threadIdx.x + s];
        __syncthreads();
    }
    if (threadIdx.x == 0) {
        colmax[c] = mx;
        colse[c]  = logf(sdata[0]);
    }
}

// ---------------------------------------------------------------------------
// NLL accumulation: acc[0] += sum_i (logp[i, labels[i]] * mask[i]),
//                   acc[1] += sum_i mask[i].
// ---------------------------------------------------------------------------
__global__ void gcn_loss_accum_kernel(const float* __restrict__ rep,
                                      const int* __restrict__ labels,
                                      const unsigned char* __restrict__ mask,
                                      const float* __restrict__ colmax,
                                      const float* __restrict__ colse,
                                      float* __restrict__ acc,
                                      int nrows) {
    __shared__ float sp[256];
    __shared__ float sm[256];
    const int i = blockIdx.x * blockDim.x + threadIdx.x;
    float p = 0.0f, mk = 0.0f;
    if (i < nrows) {
        const int l = labels[i];
        mk = mask[i] ? 1.0f : 0.0f;
        p  = (rep[(size_t)i * N_CLASSES + l] - colmax[l] - colse[l]) * mk;
    }
    sp[threadIdx.x] = p;
    sm[threadIdx.x] = mk;
    __syncthreads();
    for (int s = 128; s > 0; s >>= 1) {
        if (threadIdx.x < s) {
            sp[threadIdx.x] += sp[threadIdx.x + s];
            sm[threadIdx.x] += sm[threadIdx.x + s];
        }
        __syncthreads();
    }
    if (threadIdx.x == 0) {
        atomicAdd(&acc[0], sp[0]);
        atomicAdd(&acc[1], sm[0]);
    }
}

__global__ void gcn_finalize_kernel(const float* __restrict__ acc,
                                    float* __restrict__ loss_out) {
    if (threadIdx.x == 0 && blockIdx.x == 0)
        loss_out[0] = -acc[0] / acc[1];
}

// ---------------------------------------------------------------------------
// Orchestration
// ---------------------------------------------------------------------------
extern "C" void kernel_launch(void* const* d_in, const int* in_sizes, int n_in,
                              void* d_out, int out_size, void* d_ws, size_t ws_size,
                              hipStream_t stream) {
    const float* feat = (const float*)d_in[0];
    const float* W0   = (const float*)d_in[1];
    const float* b0   = (const float*)d_in[2];
    const float* W1   = (const float*)d_in[3];
    const float* b1   = (const float*)d_in[4];
    const float* W2   = (const float*)d_in[5];
    const float* b2   = (const float*)d_in[6];
    const int*   src  = (const int*)d_in[7];
    const int*   dst  = (const int*)d_in[8];
    const int*   lab  = (const int*)d_in[9];
    const unsigned char* mask = (const unsigned char*)d_in[10];

    const int nEdges = in_sizes[7];
    const int N = N_NODES;

    // Workspace layout: two NxH fp32 ping-pong buffers + reduction scalars.
    float* A      = (float*)d_ws;                   // agg buffer  [N x 64]
    float* B      = A + (size_t)N * N_HIDDEN;       // hidden buf  [N x 64]
    float* colmax = B + (size_t)N * N_HIDDEN;       // [64] (40 used)
    float* colse  = colmax + 64;                    // [64] (40 used)
    float* acc    = colse + 64;                     // [2]

    float* rep  = (float*)d_out;                    // [N x 40]
    float* loss = rep + (size_t)N * N_CLASSES;      // [1]

    const int threads = 256;
    const long long aggN   = (long long)N * N_HIDDEN;
    const int zeroBlocks   = 2048;
    const int scatBlocks   = (int)(((long long)nEdges * 16 + threads - 1) / threads);
    const int mtiles       = (N + 15) / 16;
    const int gemmBlocks   = (mtiles + 7) / 8;      // 8 wave32 waves per block
    const int nodeBlocks   = (N + threads - 1) / threads;

    // Layer 0: agg = scatter(features); h0 = relu(agg @ W0 + b0)
    gcn_zero_kernel<<<zeroBlocks, threads, 0, stream>>>(A, aggN);
    gcn_scatter_add_kernel<<<scatBlocks, threads, 0, stream>>>(feat, src, dst, A, nEdges);
    gcn_gemm_wmma_kernel<N_HIDDEN, true><<<gemmBlocks, threads, 0, stream>>>(A, W0, b0, B, N);

    // Layer 1
    gcn_zero_kernel<<<zeroBlocks, threads, 0, stream>>>(A, aggN);
    gcn_scatter_add_kernel<<<scatBlocks, threads, 0, stream>>>(B, src, dst, A, nEdges);
    gcn_gemm_wmma_kernel<N_HIDDEN, true><<<gemmBlocks, threads, 0, stream>>>(A, W1, b1, B, N);

    // Layer 2 (no activation) -> rep in d_out
    gcn_zero_kernel<<<zeroBlocks, threads, 0, stream>>>(A, aggN);
    gcn_scatter_add_kernel<<<scatBlocks, threads, 0, stream>>>(B, src, dst, A, nEdges);
    gcn_gemm_wmma_kernel<N_CLASSES, false><<<gemmBlocks, threads, 0, stream>>>(A, W2, b2, rep, N);

    // Loss: log_softmax over axis=0, NLL mean over mask.
    gcn_zero_kernel<<<1, threads, 0, stream>>>(colmax, 130);  // colmax+colse+acc
    gcn_col_logsumexp_kernel<<<N_CLASSES, threads, 0, stream>>>(rep, N, colmax, colse);
    gcn_loss_accum_kernel<<<nodeBlocks, threads, 0, stream>>>(rep, lab, mask, colmax, colse, acc, N);
    gcn_finalize_kernel<<<1, 1, 0, stream>>>(acc, loss);
}